// Embedding_23141283791086
// MI455X (gfx1250) — compile-verified
//
#include <hip/hip_runtime.h>

// out[i,j,k] = C[i,j,k] * x[i,j]
// B=4096, INPUT_SIZE=512, EMBED_DIM=64  ->  134,217,728 f32 elements out.
// Pure bandwidth-bound stream (AI ~= 0.125 flop/byte): optimal MI455X kernel
// is 128-bit coalesced loads/stores with non-temporal hints on the 1 GiB
// C/out streams (they exceed the 192 MB L2 and have zero reuse), while the
// 8 MiB x table uses default caching for its 16x reuse in L2.

typedef __attribute__((ext_vector_type(4))) float v4f;

__global__ __launch_bounds__(256) void scale_rows_b128_nt(
    const v4f* __restrict__ C4,      // C viewed as float4: 16 float4 per row
    const float* __restrict__ x,     // one scale per 64-elem row
    v4f* __restrict__ out4,
    unsigned int n4)                 // total float4 count = 33,554,432
{
    unsigned int i = blockIdx.x * blockDim.x + threadIdx.x;
    if (i >= n4) return;

    // 64 floats per (b, j) row => 16 float4 chunks per row.
    float s = x[i >> 4];                           // RT-cached, 16x reuse in L2

    v4f c = __builtin_nontemporal_load(C4 + i);    // global_load_b128, TH=NT
    v4f r = c * s;
    __builtin_nontemporal_store(r, out4 + i);      // global_store_b128, TH=NT
}

extern "C" void kernel_launch(void* const* d_in, const int* in_sizes, int n_in,
                              void* d_out, int out_size, void* d_ws, size_t ws_size,
                              hipStream_t stream) {
    (void)in_sizes; (void)n_in; (void)d_ws; (void)ws_size;

    const float* x  = (const float*)d_in[0];   // (4096, 512) f32
    const v4f*   C4 = (const v4f*)  d_in[1];   // (4096, 512, 64) f32 as float4
    v4f*         o4 = (v4f*)        d_out;

    unsigned int n4 = (unsigned int)(out_size / 4);   // 33,554,432 float4s
    const unsigned int block = 256;
    unsigned int grid = (n4 + block - 1) / block;     // 131,072 blocks

    scale_rows_b128_nt<<<grid, block, 0, stream>>>(C4, x, o4, n4);
}